// SparseCayleyStringPE_33380485824721
// MI455X (gfx1250) — compile-verified
//
#include <hip/hip_runtime.h>
#include <hip/hip_bf16.h>
#include <math.h>
#include <stdint.h>

#define D_MODEL 256
#define D_HALF  128
#define CG_ITERS 20
#define CG_TOL   1e-5f

typedef __attribute__((ext_vector_type(8))) float v8f;
typedef __attribute__((ext_vector_type(2))) float v2f;
typedef __attribute__((ext_vector_type(4))) unsigned int v4u;
typedef __attribute__((ext_vector_type(8))) int v8i;
typedef __attribute__((ext_vector_type(4))) int v4i;

// Swizzled B-operand layout: element (k,n) of a 256x256 matrix lives at
// (k>>1)*512 + 2n + (k&1), so rows k,k+1 at column n form a contiguous
// 8B-aligned float2 -> single ds_load_b64 per WMMA B operand.
__host__ __device__ __forceinline__ int swz(int k, int n) {
  return (k >> 1) * (2 * D_MODEL) + 2 * n + (k & 1);
}

// -------- butterfly reduction over a 16-lane half of the wave32 --------
__device__ __forceinline__ float hred16(float x) {
#pragma unroll
  for (int m = 1; m < 16; m <<= 1) x += __shfl_xor(x, m, 16);
  return x;
}

// ---------------------------------------------------------------------
// TDM: DMA a contiguous 256KB (256x256 f32) block from global into LDS.
// D# group0: count=1, lds_addr, global_addr[56:0], type=2.
// D# group1: data_size=4B, tensor_dim0/1=256, tile_dim0/1=256,
//            tensor_dim0_stride=256 (elements).
// ---------------------------------------------------------------------
__device__ __forceinline__ void tdm_stage_256x256(unsigned lds_off,
                                                  const float* gsrc) {
  unsigned long long ga = (unsigned long long)(uintptr_t)gsrc;
  v4u g0;
  g0[0] = 1u;                                    // count=1, user descriptor
  g0[1] = lds_off;                               // lds_addr (bytes)
  g0[2] = (unsigned)ga;                          // global_addr[31:0]
  g0[3] = (unsigned)((ga >> 32) & 0x01FFFFFFu)   // global_addr[56:32]
          | 0x80000000u;                         // type=2 ("image")
  v8i g1;
  g1[0] = 0x00020000;                            // data_size=2 (4 bytes)
  g1[1] = (int)(256u << 16);                     // tensor_dim0[15:0] @ [63:48]
  g1[2] = (int)(256u << 16);                     // tensor_dim1[15:0] @ [95:80]
  g1[3] = (int)(256u << 16);                     // tile_dim0 @ [127:112]
  g1[4] = 256;                                   // tile_dim1 @ [143:128]
  g1[5] = 256;                                   // tensor_dim0_stride @ [191:160]
  g1[6] = 0;
  g1[7] = 0;
  v4i gz4 = {0, 0, 0, 0};
  v8i gz8 = {0, 0, 0, 0, 0, 0, 0, 0};
  __builtin_amdgcn_tensor_load_to_lds(g0, g1, gz4, gz4, gz8, 0);
}

// ---------------------------------------------------------------------
// 16x256 tile (row-major in LDS bounce) times 256x256 matrix resident in
// LDS in swizzled layout.  Result in WMMA C-layout:
//   acc[j][v] = element(row = v + 8*(lane>=16), col = 16*j + (lane&15)).
// ---------------------------------------------------------------------
__device__ __forceinline__ void gemm_tile(const float* lds_src,
                                          const float* lds_M,
                                          int mA, int koff, int khalf,
                                          int nB, v8f acc[16]) {
  v8f zv;
#pragma unroll
  for (int e = 0; e < 8; ++e) zv[e] = 0.0f;
#pragma unroll
  for (int j = 0; j < 16; ++j) acc[j] = zv;

  for (int kc = 0; kc < 64; ++kc) {
    v2f a = *(const v2f*)(lds_src + mA * D_MODEL + 4 * kc + koff);
    const float* bp = lds_M + (2 * kc + khalf) * (2 * D_MODEL) + 2 * nB;
#pragma unroll
    for (int j = 0; j < 16; ++j) {
      v2f b = *(const v2f*)(bp + 32 * j);
      acc[j] = __builtin_amdgcn_wmma_f32_16x16x4_f32(
          false, a, false, b, (short)0, acc[j], false, false);
    }
  }
}

// ------- prep kernels: dense S (plain), Tsw = I - S, Asw = I - S*S -----
// T and A are written in the swizzled B-operand layout.
__global__ void k_init_ST(float* __restrict__ S, float* __restrict__ Tsw) {
  int i = blockIdx.x, j = threadIdx.x;
  S[i * D_MODEL + j] = 0.0f;
  Tsw[swz(i, j)] = (i == j) ? 1.0f : 0.0f;
}

__global__ void k_scatter(const int* __restrict__ rows,
                          const int* __restrict__ cols,
                          const float* __restrict__ sv, int nnz,
                          float* __restrict__ S, float* __restrict__ Tsw) {
  for (int e = threadIdx.x; e < nnz; e += blockDim.x) {
    int r = rows[e], c = cols[e];
    float s = sv[e];
    // smv(X) = X*S with S[c][r] = +s, S[r][c] = -s  (r < c, pairs unique)
    S[c * D_MODEL + r] = s;
    S[r * D_MODEL + c] = -s;
    Tsw[swz(c, r)] = -s;  // T = I - S
    Tsw[swz(r, c)] = s;
  }
}

__global__ void k_build_A(const float* __restrict__ S, float* __restrict__ Asw) {
  int i = blockIdx.x, j = threadIdx.x;
  float acc = 0.0f;
  for (int kk = 0; kk < D_MODEL; ++kk)
    acc += S[i * D_MODEL + kk] * S[kk * D_MODEL + j];
  Asw[swz(i, j)] = ((i == j) ? 1.0f : 0.0f) - acc;
}

// ---------------------------------------------------------------------
// Fused: RoPE -> b = h*T -> 20 CG iterations on A = I - S^2 -> Y = z*T.
// 4 wave32s per block, 16 rows per wave.  p, r, Ap, z in VGPRs
// (WMMA C-layout).  LDS: 256KB staged matrix (TDM) + 4x16KB wave-private
// transpose bounce = 320KB.  The iteration loop is barrier-free.
// ---------------------------------------------------------------------
__global__ void __launch_bounds__(128) cayley_cg_fused(
    const float* __restrict__ q, const float* __restrict__ kmat,
    const float* __restrict__ freqs, const int* __restrict__ pos,
    const float* __restrict__ Tsw, const float* __restrict__ Asw,
    float* __restrict__ out, int qRows, int N) {
  extern __shared__ float smem[];
  float* smemM = smem;  // 256x256 staged matrix (swizzled)
  const int tid = threadIdx.x;
  const int lane = tid & 31;
  const int wid = tid >> 5;
  float* bounce = smem + 65536 + wid * (16 * D_MODEL);  // wave-private 16x256

  const int rowbase = blockIdx.x * 64 + wid * 16;
  const int mA = lane & 15;
  const int khalf = lane >> 4;  // 0 or 1
  const int koff = khalf << 1;  // 0 or 2
  const int nB = lane & 15;
  const int h = lane >> 4;

  const unsigned ldsM = (unsigned)(uintptr_t)smemM;

  // ---- start DMA of T while we do RoPE ----
  if (wid == 0) tdm_stage_256x256(ldsM, Tsw);

  // ---- RoPE the wave's 16 rows into its bounce buffer ----
  for (int t = lane; t < 16 * D_HALF; t += 32) {
    int m = t >> 7;            // row within tile
    int d = t & (D_HALF - 1);  // frequency pair index
    int row = rowbase + m;
    const float* src = (row < qRows)
                           ? (q + (size_t)row * D_MODEL)
                           : (kmat + (size_t)(row - qRows) * D_MODEL);
    int n = row % N;
    float ang = (float)pos[n] * freqs[d];
    float sn = sinf(ang), cs = cosf(ang);
    v2f xy = *(const v2f*)(src + 2 * d);
    bounce[m * D_MODEL + 2 * d] = xy.x * cs - xy.y * sn;
    bounce[m * D_MODEL + 2 * d + 1] = xy.x * sn + xy.y * cs;
  }
  if (wid == 0) __builtin_amdgcn_s_wait_tensorcnt(0);
  __syncthreads();  // T staged, bounces ready

  // ---- b = H_tile * T ;  r = p = b ;  z = 0 ----
  v8f p[16], r[16], z[16], Ap[16];
  gemm_tile(bounce, smemM, mA, koff, khalf, nB, p);
#pragma unroll
  for (int j = 0; j < 16; ++j) {
    r[j] = p[j];
#pragma unroll
    for (int e = 0; e < 8; ++e) z[j][e] = 0.0f;
  }

  __syncthreads();  // everyone done reading T from LDS
  if (wid == 0) tdm_stage_256x256(ldsM, Asw);  // DMA A, overlapped with rsold

  float rsold[8];
  bool dn[8];
#pragma unroll
  for (int v = 0; v < 8; ++v) {
    float rs = 0.0f;
#pragma unroll
    for (int j = 0; j < 16; ++j) rs += r[j][v] * r[j][v];
    rsold[v] = hred16(rs);
    dn[v] = false;
  }
  if (wid == 0) __builtin_amdgcn_s_wait_tensorcnt(0);
  __syncthreads();  // A staged

  // ---- CG iterations (barrier-free: bounce is wave-private, M read-only)
  for (int it = 0; it < CG_ITERS; ++it) {
    // transpose-bounce p: C-layout registers -> row-major LDS
#pragma unroll
    for (int j = 0; j < 16; ++j)
#pragma unroll
      for (int v = 0; v < 8; ++v)
        bounce[(v + 8 * h) * D_MODEL + 16 * j + nB] = p[j][v];

    gemm_tile(bounce, smemM, mA, koff, khalf, nB, Ap);  // Ap = p*(I - S^2)

#pragma unroll
    for (int v = 0; v < 8; ++v) {
      float pap = 0.0f;
#pragma unroll
      for (int j = 0; j < 16; ++j) pap += p[j][v] * Ap[j][v];
      pap = hred16(pap);
      float alpha = dn[v] ? 0.0f : rsold[v] / (pap + 1e-12f);
      float rs = 0.0f;
#pragma unroll
      for (int j = 0; j < 16; ++j) {
        z[j][v] += alpha * p[j][v];
        r[j][v] -= alpha * Ap[j][v];
        rs += r[j][v] * r[j][v];
      }
      rs = hred16(rs);
      float rsnew = dn[v] ? rsold[v] : rs;
      bool nd = dn[v] || (sqrtf(rsnew) < CG_TOL);
      float beta = rsnew / (rsold[v] + 1e-12f);
#pragma unroll
      for (int j = 0; j < 16; ++j)
        p[j][v] = nd ? p[j][v] : (r[j][v] + beta * p[j][v]);
      rsold[v] = rsnew;
      dn[v] = nd;
    }
  }

  // ---- restage T, overlapped with bouncing z; then Y = z * T ----
  __syncthreads();  // all waves done reading A
  if (wid == 0) tdm_stage_256x256(ldsM, Tsw);
#pragma unroll
  for (int j = 0; j < 16; ++j)
#pragma unroll
    for (int v = 0; v < 8; ++v)
      bounce[(v + 8 * h) * D_MODEL + 16 * j + nB] = z[j][v];
  if (wid == 0) __builtin_amdgcn_s_wait_tensorcnt(0);
  __syncthreads();  // T staged

  gemm_tile(bounce, smemM, mA, koff, khalf, nB, Ap);
#pragma unroll
  for (int j = 0; j < 16; ++j)
#pragma unroll
    for (int v = 0; v < 8; ++v)
      out[(size_t)(rowbase + v + 8 * h) * D_MODEL + 16 * j + nB] = Ap[j][v];
}

// ---------------------------------------------------------------------
extern "C" void kernel_launch(void* const* d_in, const int* in_sizes, int n_in,
                              void* d_out, int out_size, void* d_ws,
                              size_t ws_size, hipStream_t stream) {
  const float* q = (const float*)d_in[0];
  const float* k = (const float*)d_in[1];
  const float* freqs = (const float*)d_in[2];
  const float* sv = (const float*)d_in[3];
  const int* pos = (const int*)d_in[4];
  const int* rows = (const int*)d_in[5];
  const int* cols = (const int*)d_in[6];

  const int nnz = in_sizes[3];
  const int qRows = in_sizes[0] / D_MODEL;  // B*N = 8192
  const int N = in_sizes[4];                // 4096
  const int totalRows = 2 * qRows;          // 16384

  float* S = (float*)d_ws;             // 256x256 plain
  float* Tsw = S + D_MODEL * D_MODEL;  // I - S   (swizzled)
  float* Asw = Tsw + D_MODEL * D_MODEL;  // I - S^2 (swizzled)

  k_init_ST<<<D_MODEL, D_MODEL, 0, stream>>>(S, Tsw);
  k_scatter<<<1, 256, 0, stream>>>(rows, cols, sv, nnz, S, Tsw);
  k_build_A<<<D_MODEL, D_MODEL, 0, stream>>>(S, Asw);

  const int blocks = totalRows / 64;  // 256 blocks of 4 wave32s
  const size_t shmem = (65536u + 4u * 16u * D_MODEL) * sizeof(float);  // 320KB
  cayley_cg_fused<<<blocks, 128, shmem, stream>>>(q, k, freqs, pos, Tsw, Asw,
                                                  (float*)d_out, qRows, N);
}